// GCN__54606214201441
// MI455X (gfx1250) — compile-verified
//
#include <hip/hip_runtime.h>
#include <hip/hip_bf16.h>

typedef __attribute__((ext_vector_type(2))) float v2f;
typedef __attribute__((ext_vector_type(8))) float v8f;

// ---------------------------------------------------------------------------
// Dense GEMM with bias:  Y[M x C] = X[M x K] @ W[K x C] + b,  f32 WMMA.
// One wave (32 lanes) computes one 16x16 tile of Y via V_WMMA_F32_16X16X4_F32.
// blockDim = (32, ceil(C/16)) : threadIdx.y selects the column tile.
// RELU_IN applies relu to X on load (fusing the activation of the previous
// spmm output into this GEMM's A-operand read).
// Column guarding is compile-time: when C % 16 == 0 (layers 1 and 2) no mask
// or index clamp is emitted at all; only the C=40 layer pays for it.
// ---------------------------------------------------------------------------
template <int K, int C, bool RELU_IN>
__global__ __launch_bounds__(128) void gemm_bias_wmma(
    const float* __restrict__ X, const float* __restrict__ W,
    const float* __restrict__ bias, float* __restrict__ Y, int M) {
  const int lane  = threadIdx.x;        // 0..31
  const int l15   = lane & 15;
  const int khalf = lane >> 4;          // 0 or 1
  const int m0    = blockIdx.x * 16;
  const int n0    = threadIdx.y * 16;

  // A-operand row (clamped so every lane issues a valid load; EXEC stays all 1s)
  const int m  = m0 + l15;
  const int mc = (m < M) ? m : (M - 1);
  const float* __restrict__ xrow = X + (size_t)mc * K;

  // B-operand column
  const int n = n0 + l15;
  int   nc;
  float nmask;
  if constexpr (C % 16 == 0) {
    nc    = n;       // always in range
    nmask = 1.0f;    // unused; folded away
  } else {
    nc    = (n < C) ? n : (C - 1);
    nmask = (n < C) ? 1.0f : 0.0f;
  }

  v8f acc = {};
#pragma unroll
  for (int k = 0; k < K; k += 4) {
    const int ka = k + 2 * khalf;
    v2f a, b;
    float a0 = xrow[ka + 0];
    float a1 = xrow[ka + 1];
    if (RELU_IN) {
      a0 = fmaxf(a0, 0.0f);
      a1 = fmaxf(a1, 0.0f);
    }
    a.x = a0;
    a.y = a1;
    if constexpr (C % 16 == 0) {
      b.x = W[(size_t)(ka + 0) * C + nc];
      b.y = W[(size_t)(ka + 1) * C + nc];
    } else {
      b.x = W[(size_t)(ka + 0) * C + nc] * nmask;
      b.y = W[(size_t)(ka + 1) * C + nc] * nmask;
    }
    // D = A x B + C    (neg_a, A, neg_b, B, c_mod, C, reuse_a, reuse_b)
    acc = __builtin_amdgcn_wmma_f32_16x16x4_f32(false, a, false, b, (short)0,
                                                acc, false, false);
  }

  if (n < C) {
    const float bv = bias[n];
#pragma unroll
    for (int r = 0; r < 8; ++r) {
      const int mr = m0 + r + 8 * khalf;   // C/D layout: VGPR r -> row r + 8*hi
      if (mr < M) Y[(size_t)mr * C + n] = acc[r] + bv;
    }
  }
}

// ---------------------------------------------------------------------------
// SpMM scatter: Y[rows[e], :] += vals[e] * X[cols[e], :]
// Activation matrices (25.6 MB) live in L2 (192 MB), so the random gathers
// and f32 atomics resolve on-chip. float4 gathers -> global_load_b128,
// accumulation via global_atomic_add_f32 (no return -> STOREcnt path).
// blockDim = (C/4, EDGES_PER_BLOCK)
// ---------------------------------------------------------------------------
template <int C>
__global__ void spmm_scatter(const int* __restrict__ rows,
                             const int* __restrict__ cols,
                             const float* __restrict__ vals,
                             const float* __restrict__ X,
                             float* __restrict__ Y, int n_edges) {
  const int e = blockIdx.x * blockDim.y + threadIdx.y;
  if (e >= n_edges) return;
  const int   r = rows[e];
  const int   c = cols[e];
  const float v = vals[e];
  const int   f4 = threadIdx.x;  // 0 .. C/4-1
  const float4 x =
      *(const float4*)(X + (size_t)c * C + (size_t)f4 * 4);
  float* yp = Y + (size_t)r * C + (size_t)f4 * 4;
  atomicAdd(yp + 0, v * x.x);
  atomicAdd(yp + 1, v * x.y);
  atomicAdd(yp + 2, v * x.z);
  atomicAdd(yp + 3, v * x.w);
}

// ---------------------------------------------------------------------------
// Cross-entropy: mean over nodes of (logsumexp(logits) - logits[label]).
// One thread per node, LDS block reduction, atomic accumulate of partial/N.
// ---------------------------------------------------------------------------
template <int C>
__global__ __launch_bounds__(256) void ce_loss(const float* __restrict__ H,
                                               const int* __restrict__ labels,
                                               float* __restrict__ out,
                                               int n_nodes) {
  const int i = blockIdx.x * blockDim.x + threadIdx.x;
  float nll = 0.0f;
  if (i < n_nodes) {
    const float* l = H + (size_t)i * C;
    float mx = l[0];
#pragma unroll
    for (int c = 1; c < C; ++c) mx = fmaxf(mx, l[c]);
    float s = 0.0f;
#pragma unroll
    for (int c = 0; c < C; ++c) s += __expf(l[c] - mx);
    const float lse = mx + __logf(s);
    nll = lse - l[labels[i]];
  }
  __shared__ float red[256];
  red[threadIdx.x] = nll;
  __syncthreads();
#pragma unroll
  for (int off = 128; off > 0; off >>= 1) {
    if (threadIdx.x < off) red[threadIdx.x] += red[threadIdx.x + off];
    __syncthreads();
  }
  if (threadIdx.x == 0) atomicAdd(out, red[0] * (1.0f / (float)n_nodes));
}

extern "C" void kernel_launch(void* const* d_in, const int* in_sizes, int n_in,
                              void* d_out, int out_size, void* d_ws,
                              size_t ws_size, hipStream_t stream) {
  (void)n_in; (void)out_size; (void)ws_size;
  // setup_inputs() order:
  const float* features = (const float*)d_in[0];   // N x 128
  const int*   labels   = (const int*)d_in[1];     // N
  const int*   rows     = (const int*)d_in[2];     // E
  const int*   cols     = (const int*)d_in[3];     // E
  const float* vals     = (const float*)d_in[4];   // E
  const float* W1       = (const float*)d_in[5];   // 128 x 64
  const float* b1       = (const float*)d_in[6];   // 64
  const float* W2       = (const float*)d_in[7];   // 64 x 64
  const float* b2       = (const float*)d_in[8];   // 64
  const float* W3       = (const float*)d_in[9];   // 64 x 40
  const float* b3       = (const float*)d_in[10];  // 40

  const int N = in_sizes[1];  // 100000 nodes
  const int E = in_sizes[2];  // 1600000 edges

  constexpr int KIN = 128, H = 64, CL = 40;

  // Workspace layout (f32)
  float* XW  = (float*)d_ws;              // N*64   : features @ W1 + b1
  float* H1  = XW  + (size_t)N * H;       // N*64   : spmm output 1 (zeroed)
  float* HW2 = H1  + (size_t)N * H;       // N*64   : relu(H1) @ W2 + b2
  float* H2  = HW2 + (size_t)N * H;       // N*64   : spmm output 2 (zeroed)
  float* HW3 = H2  + (size_t)N * H;       // N*40   : relu(H2) @ W3 + b3
  float* H3  = HW3 + (size_t)N * CL;      // N*40   : spmm output 3 (zeroed)

  // Zero accumulation targets + scalar output (graph-capture safe).
  hipMemsetAsync(H1, 0, (size_t)N * H  * sizeof(float), stream);
  hipMemsetAsync(H2, 0, (size_t)N * H  * sizeof(float), stream);
  hipMemsetAsync(H3, 0, (size_t)N * CL * sizeof(float), stream);
  hipMemsetAsync(d_out, 0, sizeof(float), stream);

  const int mtiles = (N + 15) / 16;

  // Layer 1: XW = features @ W1 + b1 ; H1 = spmm(XW)
  gemm_bias_wmma<KIN, H, false>
      <<<dim3(mtiles), dim3(32, H / 16), 0, stream>>>(features, W1, b1, XW, N);
  {
    dim3 blk(H / 4, 16);  // 16 float4s/edge-thread-row, 16 edges/block = 256 thr
    dim3 grd((E + blk.y - 1) / blk.y);
    spmm_scatter<H><<<grd, blk, 0, stream>>>(rows, cols, vals, XW, H1, E);
  }

  // Layer 2: HW2 = relu(H1) @ W2 + b2 ; H2 = spmm(HW2)
  gemm_bias_wmma<H, H, true>
      <<<dim3(mtiles), dim3(32, H / 16), 0, stream>>>(H1, W2, b2, HW2, N);
  {
    dim3 blk(H / 4, 16);
    dim3 grd((E + blk.y - 1) / blk.y);
    spmm_scatter<H><<<grd, blk, 0, stream>>>(rows, cols, vals, HW2, H2, E);
  }

  // Layer 3: HW3 = relu(H2) @ W3 + b3 ; H3 = spmm(HW3)
  gemm_bias_wmma<H, CL, true>
      <<<dim3(mtiles), dim3(32, (CL + 15) / 16), 0, stream>>>(H2, W3, b3, HW3, N);
  {
    dim3 blk(CL / 4, 25);  // 10 x 25 = 250 threads
    dim3 grd((E + blk.y - 1) / blk.y);
    spmm_scatter<CL><<<grd, blk, 0, stream>>>(rows, cols, vals, HW3, H3, E);
  }

  // Mean cross-entropy over nodes -> scalar
  ce_loss<CL><<<dim3((N + 255) / 256), dim3(256), 0, stream>>>(
      H3, labels, (float*)d_out, N);
}